// SLAgent_24816321036736
// MI455X (gfx1250) — compile-verified
//
#include <hip/hip_runtime.h>

typedef float v2f __attribute__((ext_vector_type(2)));
typedef float v8f __attribute__((ext_vector_type(8)));

#define NW 4  // waves per block (128 threads)

// One 16(M) x 16(N) fp32 tile:  D = A[16xK] * B[Kx16] + bias(broadcast per column)
// A: row-major in LDS with stride sa (K padded to multiple of 4 with zeros)
// B: row-major (K x n) with leading dim ldb; caller pre-offsets column via `col`
// fp32 WMMA operand layout (ISA 7.12.2):
//   A: lane = half*16+row holds (row, kb+2*half) in v[0], (row, kb+2*half+1) in v[1]
//   B: lane = half*16+col holds (kb+2*half, col) in v[0], (kb+2*half+1, col) in v[1]
//   D: vgpr j on lane half*16+col = (j + 8*half, col)
static __device__ __forceinline__ v8f tile_mm(const float* As, int sa,
                                              const float* Bs, int ldb,
                                              int K, int row, int half, int col,
                                              float bias) {
  v8f acc;
#pragma unroll
  for (int j = 0; j < 8; ++j) acc[j] = bias;
#pragma unroll
  for (int kb = 0; kb < K; kb += 4) {
    int ka = kb + 2 * half;
    v2f a, b;
    a[0] = As[row * sa + ka];
    a[1] = As[row * sa + ka + 1];
    b[0] = Bs[ka * ldb + col];
    b[1] = Bs[(ka + 1) * ldb + col];
    acc = __builtin_amdgcn_wmma_f32_16x16x4_f32(false, a, false, b,
                                                (short)0, acc, false, false);
  }
  return acc;
}

struct WaveScratch {
  float obs[16 * 12];  // obs tile, K padded 10->12 with zeros
  float h1[16 * 34];   // z layer1 out [16][32], stride 34
  float h2[16 * 18];   // z layer2 out [16][16], stride 18
  float z[16 * 17];    // gathered z_logits [16][16], stride 17
  float vs[16 * 33];   // trunk out [16][32], stride 33
  int   ty[16];        // type ids of the 16 samples
  int   bm[16];        // best mode per sample
};

__global__ __launch_bounds__(NW * 32)
void slagent_fused_kernel(const float* __restrict__ obs,
                          const int*   __restrict__ type_ids,
                          const float* __restrict__ eps,
                          const float* __restrict__ W_root,
                          const float* __restrict__ b_root,
                          const float* __restrict__ Wz1,
                          const float* __restrict__ bz1,
                          const float* __restrict__ Wz2,
                          const float* __restrict__ bz2,
                          const float* __restrict__ Wz3,
                          const float* __restrict__ bz3,
                          const float* __restrict__ Wx1,
                          const float* __restrict__ bx1,
                          const float* __restrict__ Wx2,
                          const float* __restrict__ bx2,
                          const float* __restrict__ Wy1,
                          const float* __restrict__ by1,
                          const float* __restrict__ Wy2,
                          const float* __restrict__ by2,
                          float* __restrict__ out, long long Ntot) {
  __shared__ float Wr_s[12 * 33];       // W_root, K padded 10->12 (zeros)
  __shared__ float Wz1_s[3 * 12 * 33];  // Wz1 per type, K padded
  __shared__ WaveScratch wsc[NW];

  const int tid = threadIdx.x;
  // Stage K-padded weights once per block.
  for (int idx = tid; idx < 12 * 33; idx += NW * 32) {
    int r = idx / 33, c = idx - r * 33;
    Wr_s[idx] = (r < 10 && c < 32) ? W_root[r * 32 + c] : 0.0f;
  }
  for (int idx = tid; idx < 3 * 12 * 33; idx += NW * 32) {
    int t = idx / 396, rem = idx - t * 396;
    int r = rem / 33, c = rem - r * 33;
    Wz1_s[idx] = (r < 10 && c < 32) ? Wz1[t * 320 + r * 32 + c] : 0.0f;
  }
  __syncthreads();

  const int wave = tid >> 5, lane = tid & 31;
  const int half = lane >> 4, row = lane & 15;
  WaveScratch& w = wsc[wave];
  const long long tileBase = ((long long)blockIdx.x * NW + wave) * 16;

  // Load 16 samples of obs into LDS (zero-pad k=10,11); load type ids.
  for (int idx = lane; idx < 16 * 12; idx += 32) {
    int r = idx / 12, c = idx - r * 12;
    w.obs[idx] = (c < 10) ? obs[(tileBase + r) * 10 + c] : 0.0f;
  }
  if (lane < 16) w.ty[lane] = type_ids[tileBase + lane];

  // ---- z-MLP, dense over the 3 types, gather by each sample's type ----
  for (int t = 0; t < 3; ++t) {
    const float* Wz1t = Wz1_s + t * 396;
    // layer1: [16x10] x [10x32] -> relu -> h1
#pragma unroll
    for (int n0 = 0; n0 < 32; n0 += 16) {
      v8f d = tile_mm(w.obs, 12, Wz1t, 33, 12, row, half, n0 + row,
                      bz1[t * 32 + n0 + row]);
#pragma unroll
      for (int j = 0; j < 8; ++j)
        w.h1[(j + 8 * half) * 34 + n0 + row] = fmaxf(d[j], 0.0f);
    }
    // layer2: [16x32] x [32x16] -> relu -> h2   (B straight from global, K%4==0)
    {
      v8f d = tile_mm(w.h1, 34, Wz2 + t * 512, 16, 32, row, half, row,
                      bz2[t * 16 + row]);
#pragma unroll
      for (int j = 0; j < 8; ++j)
        w.h2[(j + 8 * half) * 18 + row] = fmaxf(d[j], 0.0f);
    }
    // layer3: [16x16] x [16x16] -> z (no relu); gather rows whose type == t
    {
      v8f d = tile_mm(w.h2, 18, Wz3 + t * 256, 16, 16, row, half, row,
                      bz3[t * 16 + row]);
#pragma unroll
      for (int j = 0; j < 8; ++j) {
        int r2 = j + 8 * half;
        if (w.ty[r2] == t) w.z[r2 * 17 + row] = d[j];
      }
    }
  }

  // Write z_logits: contiguous 256 floats per tile, coalesced.
  {
    float* outz = out + (size_t)(2 * Ntot) + (size_t)tileBase * 16;
#pragma unroll
    for (int i = 0; i < 8; ++i) {
      int idx = lane + 32 * i;
      outz[idx] = w.z[(idx >> 4) * 17 + (idx & 15)];
    }
  }

  // argmax over 16 logits (strict > keeps first max, matching jnp.argmax)
  if (half == 0) {
    int bm = 0;
    float best = w.z[row * 17];
    for (int m = 1; m < 16; ++m) {
      float v = w.z[row * 17 + m];
      if (v > best) { best = v; bm = m; }
    }
    w.bm[row] = bm;
  }

  // ---- shared trunk: vs = relu(obs @ W_root + b_root) ----
#pragma unroll
  for (int n0 = 0; n0 < 32; n0 += 16) {
    v8f d = tile_mm(w.obs, 12, Wr_s, 33, 12, row, half, n0 + row,
                    b_root[n0 + row]);
#pragma unroll
    for (int j = 0; j < 8; ++j)
      w.vs[(j + 8 * half) * 33 + n0 + row] = fmaxf(d[j], 0.0f);
  }

  // ---- gathered per-mode expert head: lanes 0-15 -> x, lanes 16-31 -> y ----
  {
    const int s = row;
    const int mode = w.bm[s];
    const float* W1 = half ? (Wy1 + mode * 256) : (Wx1 + mode * 256);
    const float* B1 = half ? (by1 + mode * 8)   : (bx1 + mode * 8);
    const float* W2 = half ? (Wy2 + mode * 16)  : (Wx2 + mode * 16);
    const float* B2 = half ? (by2 + mode * 2)   : (bx2 + mode * 2);
    float h[8];
#pragma unroll
    for (int k = 0; k < 8; ++k) h[k] = B1[k];
    for (int hi = 0; hi < 32; ++hi) {
      float v = w.vs[s * 33 + hi];
#pragma unroll
      for (int k = 0; k < 8; ++k) h[k] = fmaf(v, W1[hi * 8 + k], h[k]);
    }
    float o0 = B2[0], o1 = B2[1];
#pragma unroll
    for (int k = 0; k < 8; ++k) {
      float hk = fmaxf(h[k], 0.0f);
      o0 = fmaf(hk, W2[2 * k], o0);
      o1 = fmaf(hk, W2[2 * k + 1], o1);
    }
    float sc = fmaxf(o1, 0.0f) + 1e-8f;
    long long g = tileBase + s;
    out[2 * g + half] = fmaf(sc, eps[2 * g + half], o0);
  }
}

extern "C" void kernel_launch(void* const* d_in, const int* in_sizes, int n_in,
                              void* d_out, int out_size, void* d_ws, size_t ws_size,
                              hipStream_t stream) {
  const float* obs      = (const float*)d_in[0];
  const int*   type_ids = (const int*)  d_in[1];
  const float* eps      = (const float*)d_in[2];
  const float* W_root   = (const float*)d_in[3];
  const float* b_root   = (const float*)d_in[4];
  const float* Wz1      = (const float*)d_in[5];
  const float* bz1      = (const float*)d_in[6];
  const float* Wz2      = (const float*)d_in[7];
  const float* bz2      = (const float*)d_in[8];
  const float* Wz3      = (const float*)d_in[9];
  const float* bz3      = (const float*)d_in[10];
  const float* Wx1      = (const float*)d_in[11];
  const float* bx1      = (const float*)d_in[12];
  const float* Wx2      = (const float*)d_in[13];
  const float* bx2      = (const float*)d_in[14];
  const float* Wy1      = (const float*)d_in[15];
  const float* by1      = (const float*)d_in[16];
  const float* Wy2      = (const float*)d_in[17];
  const float* by2      = (const float*)d_in[18];

  long long N = in_sizes[1];                 // 524288, divisible by 16*NW
  int blocks = (int)(N / (16 * NW));
  slagent_fused_kernel<<<blocks, NW * 32, 0, stream>>>(
      obs, type_ids, eps, W_root, b_root,
      Wz1, bz1, Wz2, bz2, Wz3, bz3,
      Wx1, bx1, Wx2, bx2, Wy1, by1, Wy2, by2,
      (float*)d_out, N);
}